// EwaldReciprocal_49331994362110
// MI455X (gfx1250) — compile-verified
//
#include <hip/hip_runtime.h>
#include <math.h>

typedef __attribute__((ext_vector_type(2))) float v2f;
typedef __attribute__((ext_vector_type(8))) float v8f;

#define BOHR_F 1.8897261258369282f
#define PI_F   3.14159265358979323846f

// Hardware trig: v_sin_f32/v_cos_f32 compute sin/cos(2*pi*x), so we feed the
// *fractional* phase (kvec . s) directly -- no 2*pi multiply, no range-reduction
// multiply. All 8 sins then all 8 coses in ONE asm block: every trans op except
// the last is followed by an independent trans op, so a single trailing v_nop
// covers the multicycle TRANS RAW hazard (vs. one v_nop per pair).
__device__ __forceinline__ void sincos2pi_x8(v8f x, v8f* sn, v8f* cs) {
    float s0, s1, s2, s3, s4, s5, s6, s7;
    float c0, c1, c2, c3, c4, c5, c6, c7;
    asm volatile(
        "v_sin_f32 %0, %16\n\t"
        "v_sin_f32 %1, %17\n\t"
        "v_sin_f32 %2, %18\n\t"
        "v_sin_f32 %3, %19\n\t"
        "v_sin_f32 %4, %20\n\t"
        "v_sin_f32 %5, %21\n\t"
        "v_sin_f32 %6, %22\n\t"
        "v_sin_f32 %7, %23\n\t"
        "v_cos_f32 %8, %16\n\t"
        "v_cos_f32 %9, %17\n\t"
        "v_cos_f32 %10, %18\n\t"
        "v_cos_f32 %11, %19\n\t"
        "v_cos_f32 %12, %20\n\t"
        "v_cos_f32 %13, %21\n\t"
        "v_cos_f32 %14, %22\n\t"
        "v_cos_f32 %15, %23\n\t"
        "v_nop"
        : "=&v"(s0), "=&v"(s1), "=&v"(s2), "=&v"(s3),
          "=&v"(s4), "=&v"(s5), "=&v"(s6), "=&v"(s7),
          "=&v"(c0), "=&v"(c1), "=&v"(c2), "=&v"(c3),
          "=&v"(c4), "=&v"(c5), "=&v"(c6), "=&v"(c7)
        : "v"(x[0]), "v"(x[1]), "v"(x[2]), "v"(x[3]),
          "v"(x[4]), "v"(x[5]), "v"(x[6]), "v"(x[7]));
    *sn = (v8f){s0, s1, s2, s3, s4, s5, s6, s7};
    *cs = (v8f){c0, c1, c2, c3, c4, c5, c6, c7};
}

// ---------------------------------------------------------------------------
// Kernel 0: zero the accumulators (Sri interleaved re/im, recip).
// ---------------------------------------------------------------------------
__global__ void k_zero(float* __restrict__ p, int n) {
    int i = blockIdx.x * blockDim.x + threadIdx.x;
    if (i < n) p[i] = 0.0f;
}

// ---------------------------------------------------------------------------
// Kernel 0b: pack tables for single-load inner loops.
//   packA[b*N+n] = (x, y, z, q)        -> one b128 per chunk lane (kernel 1)
//   packK[k]     = (kx, ky, kz, expfac)-> one b128 per chunk lane (kernel 2)
// ---------------------------------------------------------------------------
__global__ void k_pack(const float* __restrict__ coords, const float* __restrict__ q,
                       const float* __restrict__ kvec, const float* __restrict__ expfac,
                       float4* __restrict__ packA, float4* __restrict__ packK,
                       int BN, int NK)
{
    int i = blockIdx.x * blockDim.x + threadIdx.x;
    if (i < BN) {
        float4 p;
        p.x = coords[i * 3 + 0];
        p.y = coords[i * 3 + 1];
        p.z = coords[i * 3 + 2];
        p.w = q[i];
        packA[i] = p;
    }
    if (i < NK) {
        float4 p;
        p.x = kvec[i * 3 + 0];
        p.y = kvec[i * 3 + 1];
        p.z = kvec[i * 3 + 2];
        p.w = expfac[i];
        packK[i] = p;
    }
}

// ---------------------------------------------------------------------------
// Kernel 1: structure factors.
//   Sri[b,k] = (sum_n q*cos(2pi*kp_k.x_n), sum_n q*sin(2pi*kp_k.x_n))
// kp_k[j] = sum_i kvec[k,i]*cell_inv[i,j].  One wave owns a 16-k tile
// (A matrix, loop invariant) and streams a slice of atoms in chunks of 16
// (B matrix), one V_WMMA_F32_16X16X4_F32 per chunk.
// D layout: lane l -> atom col n0+(l&15); VGPR r -> k row kTile + r + (l>=16?8:0).
// ---------------------------------------------------------------------------
__global__ __launch_bounds__(256) void k_struct_fac(
    const float4* __restrict__ packA, const float* __restrict__ cell_inv,
    const float* __restrict__ kvec, float* __restrict__ Sri,
    int N, int NK, int splitN)
{
    const int lane  = threadIdx.x & 31;
    const int wave  = threadIdx.x >> 5;                 // 0..7
    const int kTile = (blockIdx.x * 8 + wave) * 16;
    const int b     = blockIdx.z;
    const int nPer  = N / splitN;
    const int nBeg  = blockIdx.y * nPer;

    const int  m  = lane & 15;
    const bool hi = lane >= 16;

    // A-matrix row for k = kTile + m: kp[j] = sum_i kvec[i]*cell_inv[i*3+j]
    const float kx = kvec[(kTile + m) * 3 + 0];
    const float ky = kvec[(kTile + m) * 3 + 1];
    const float kz = kvec[(kTile + m) * 3 + 2];
    const float kp0 = kx * cell_inv[0] + ky * cell_inv[3] + kz * cell_inv[6];
    const float kp1 = kx * cell_inv[1] + ky * cell_inv[4] + kz * cell_inv[7];
    const float kp2 = kx * cell_inv[2] + ky * cell_inv[5] + kz * cell_inv[8];

    v2f a;                    // 16x4 f32 A layout: lanes0-15: K=0/K=1 ; lanes16-31: K=2/K=3(=0)
    a.x = hi ? kp2 : kp0;
    a.y = hi ? 0.0f : kp1;

    v8f sumRe = {}, sumIm = {};
    const float4* pA = packA + (size_t)b * N;

#pragma unroll 2
    for (int n0 = nBeg; n0 < nBeg + nPer; n0 += 16) {
        const float4 p = pA[n0 + m];          // single b128: x,y,z,q
        v2f bm;                               // 4x16 B layout mirrors A
        bm.x = hi ? p.z : p.x;
        bm.y = hi ? 0.0f : p.y;

        v8f d = {};
        d = __builtin_amdgcn_wmma_f32_16x16x4_f32(
                false, a, false, bm, (short)0, d, false, false);

        v8f sn, cs;
        sincos2pi_x8(d, &sn, &cs);
        sumRe = sumRe + cs * p.w;             // -> v_pk_fma_f32 pairs
        sumIm = sumIm + sn * p.w;
    }

    // Reduce over the 16 atom-lanes inside each half-wave (masks < 16 keep halves apart)
#pragma unroll
    for (int r = 0; r < 8; ++r) {
        float re = sumRe[r], im = sumIm[r];
        for (int mask = 8; mask >= 1; mask >>= 1) {
            re += __shfl_xor(re, mask, 32);
            im += __shfl_xor(im, mask, 32);
        }
        sumRe[r] = re; sumIm[r] = im;
    }
    if (m == 0) {
        const int kbase = kTile + (hi ? 8 : 0);
#pragma unroll
        for (int r = 0; r < 8; ++r) {
            atomicAdd(&Sri[2 * ((size_t)b * NK + kbase + r) + 0], sumRe[r]);
            atomicAdd(&Sri[2 * ((size_t)b * NK + kbase + r) + 1], sumIm[r]);
        }
    }
}

// ---------------------------------------------------------------------------
// Kernel 2: per-atom reciprocal field.
//   recip[b,n] = sum_k expfac[k]*(Sre*cos(th) + Sim*sin(th))
// One wave owns a 16-atom tile (A = fractional coords, loop invariant) and
// streams a slice of k in chunks of 16 (B = kvec chunk).
// D layout: lane l -> k col kc+(l&15); VGPR r -> atom nTile + r + (l>=16?8:0).
// ---------------------------------------------------------------------------
__global__ __launch_bounds__(256) void k_recip(
    const float4* __restrict__ packA, const float* __restrict__ cell_inv,
    const float4* __restrict__ packK, const float2* __restrict__ Sri,
    float* __restrict__ recip, int N, int NK, int splitK)
{
    const int lane  = threadIdx.x & 31;
    const int wave  = threadIdx.x >> 5;
    const int nTile = (blockIdx.x * 8 + wave) * 16;
    const int b     = blockIdx.z;
    const int kPer  = NK / splitK;
    const int kBeg  = blockIdx.y * kPer;

    const int  m  = lane & 15;
    const bool hi = lane >= 16;

    // A-matrix row for atom n = nTile + m: s[i] = cell_inv[i,:] . x_n
    const float4 pc = packA[(size_t)b * N + nTile + m];
    const float s0 = cell_inv[0] * pc.x + cell_inv[1] * pc.y + cell_inv[2] * pc.z;
    const float s1 = cell_inv[3] * pc.x + cell_inv[4] * pc.y + cell_inv[5] * pc.z;
    const float s2 = cell_inv[6] * pc.x + cell_inv[7] * pc.y + cell_inv[8] * pc.z;

    v2f a;
    a.x = hi ? s2 : s0;
    a.y = hi ? 0.0f : s1;

    v8f acc = {};
    const float2* pS = Sri + (size_t)b * NK;

#pragma unroll 2
    for (int kc = kBeg; kc < kBeg + kPer; kc += 16) {
        const float4 kv = packK[kc + m];      // single b128: kx,ky,kz,expfac
        v2f bm;
        bm.x = hi ? kv.z : kv.x;
        bm.y = hi ? 0.0f : kv.y;

        v8f d = {};
        d = __builtin_amdgcn_wmma_f32_16x16x4_f32(
                false, a, false, bm, (short)0, d, false, false);

        const float2 S = pS[kc + m];          // single b64: (Sre, Sim)
        v8f sn, cs;
        sincos2pi_x8(d, &sn, &cs);
        acc = acc + (cs * S.x + sn * S.y) * kv.w;
    }

    // Reduce over the 16 k-lanes inside each half-wave
#pragma unroll
    for (int r = 0; r < 8; ++r) {
        float v = acc[r];
        for (int mask = 8; mask >= 1; mask >>= 1)
            v += __shfl_xor(v, mask, 32);
        acc[r] = v;
    }
    if (m == 0) {
        const int nbase = nTile + (hi ? 8 : 0);
#pragma unroll
        for (int r = 0; r < 8; ++r)
            atomicAdd(&recip[(size_t)b * N + nbase + r], acc[r]);
    }
}

// ---------------------------------------------------------------------------
// Kernel 3: epilogue  phi = recip*BOHR/(pi*V) - q*2*bewald*BOHR/sqrt(pi)
//   out[0:BN] = 0.5*q*phi ; out[BN:2BN] = phi
// ---------------------------------------------------------------------------
__global__ void k_phi(const float* __restrict__ q, const float* __restrict__ recip,
                      const float* __restrict__ volume, const float* __restrict__ bewald,
                      float* __restrict__ out, int BN)
{
    int i = blockIdx.x * blockDim.x + threadIdx.x;
    if (i >= BN) return;
    const float scale = BOHR_F / (PI_F * volume[0]);
    const float self  = 2.0f * bewald[0] * BOHR_F / sqrtf(PI_F);
    const float phi   = recip[i] * scale - q[i] * self;
    out[i]      = 0.5f * q[i] * phi;
    out[BN + i] = phi;
}

extern "C" void kernel_launch(void* const* d_in, const int* in_sizes, int n_in,
                              void* d_out, int out_size, void* d_ws, size_t ws_size,
                              hipStream_t stream)
{
    const float* coords   = (const float*)d_in[0];   // (B,N,3)
    const float* q        = (const float*)d_in[1];   // (B,N)
    const float* cell_inv = (const float*)d_in[2];   // (3,3)
    const float* kvec     = (const float*)d_in[3];   // (NK,3)
    const float* expfac   = (const float*)d_in[4];   // (NK,)
    const float* volume   = (const float*)d_in[5];   // (1,)
    const float* bewald   = (const float*)d_in[6];   // (1,)
    float* out = (float*)d_out;

    const int B  = 2;
    const int BN = in_sizes[1];
    const int N  = BN / B;
    const int NK = in_sizes[4];

    // Workspace: Sri[2*B*NK] | recip[BN] | packA[BN*4] | packK[NK*4]
    float*  Sri   = (float*)d_ws;
    float*  recip = Sri + (size_t)2 * B * NK;
    float4* packA = (float4*)(recip + BN);
    float4* packK = packA + BN;
    const size_t wsFloats = (size_t)2 * B * NK + BN + (size_t)4 * BN + (size_t)4 * NK;
    if (ws_size < wsFloats * sizeof(float)) return;

    const int SPLIT_N = 16;   // atom-slices per k-tile (kernel 1)
    const int SPLIT_K = 16;   // k-slices per atom-tile (kernel 2)

    const int zeroN = 2 * B * NK + BN;
    k_zero<<<(zeroN + 255) / 256, 256, 0, stream>>>(Sri, zeroN);

    const int packN = (BN > NK) ? BN : NK;
    k_pack<<<(packN + 255) / 256, 256, 0, stream>>>(coords, q, kvec, expfac,
                                                    packA, packK, BN, NK);

    dim3 g1(NK / 128, SPLIT_N, B);
    k_struct_fac<<<g1, 256, 0, stream>>>(packA, cell_inv, kvec, Sri, N, NK, SPLIT_N);

    dim3 g2(N / 128, SPLIT_K, B);
    k_recip<<<g2, 256, 0, stream>>>(packA, cell_inv, packK, (float2*)Sri,
                                    recip, N, NK, SPLIT_K);

    k_phi<<<(BN + 255) / 256, 256, 0, stream>>>(q, recip, volume, bewald, out, BN);
}